// lossFunc_16776142258928
// MI455X (gfx1250) — compile-verified
//
#include <hip/hip_runtime.h>
#include <hip/hip_bf16.h>
#include <cstdint>
#include <math.h>

// Problem constants (from reference): B=256, MAX_STEP=200, NUM_Q=1024
#define BB   256
#define SS   200
#define NQ   1024
#define SM1  199            // S-1 outputs per student
#define TILES 13            // ceil(199/16) 16-output tiles per student
#define KC   32             // K-chunk (floats per row per chunk)
#define WPB  4              // waves per block in phase 1

typedef __attribute__((ext_vector_type(2))) float v2f;
typedef __attribute__((ext_vector_type(8))) float v8f;
typedef int v4i __attribute__((vector_size(16)));   // pointee type of async builtin params

// ---------------- CDNA5 async global->LDS helpers ----------------
#if defined(__has_builtin)
#if __has_builtin(__builtin_amdgcn_global_load_async_to_lds_b128)
#define HAS_ASYNC_BUILTIN 1
#endif
#if __has_builtin(__builtin_amdgcn_s_wait_asynccnt)
#define HAS_WAITASYNC_BUILTIN 1
#endif
#endif

__device__ __forceinline__ void async_copy_b128(const float* gsrc, float* lds_dst) {
#if defined(HAS_ASYNC_BUILTIN)
  __builtin_amdgcn_global_load_async_to_lds_b128(
      (__attribute__((address_space(1))) v4i*)(const_cast<float*>(gsrc)),
      (__attribute__((address_space(3))) v4i*)(lds_dst),
      0, 0);
#else
  unsigned loff = (unsigned)(uintptr_t)lds_dst;   // low 32 bits of generic = LDS offset
  unsigned long long ga = (unsigned long long)(uintptr_t)gsrc;
  asm volatile("global_load_async_to_lds_b128 %0, %1, off"
               :: "v"(loff), "v"(ga) : "memory");
#endif
}

template <int N>
__device__ __forceinline__ void wait_asynccnt() {
#if defined(HAS_WAITASYNC_BUILTIN)
  __builtin_amdgcn_s_wait_asynccnt(N);
#else
  asm volatile("s_wait_asynccnt %0" :: "i"(N) : "memory");
#endif
}

// ---------------- Phase 1: WMMA dot products + diff row sums ----------------
// One wave handles one (b, 16-output tile). C = P * Delta^T via
// V_WMMA_F32_16X16X4_F32 accumulated over K=1024; diag(C) = p values.
// diff row-sums computed on the fly from the B-operand halves.
__global__ __launch_bounds__(WPB * 32) void dkt_phase1(
    const float* __restrict__ pred, const float* __restrict__ batch,
    float* __restrict__ p_ws, float* __restrict__ a_ws) {
  // per wave: double buffer * 3 streams (pred, h0, h1) * 16 rows * KC floats
  __shared__ float smem[WPB * 2 * 3 * 16 * KC];   // 48 KB
  const int lane = threadIdx.x & 31;
  const int wid  = threadIdx.x >> 5;
  const int task = blockIdx.x * WPB + wid;
  if (task >= BB * TILES) return;                 // uniform per-wave exit
  const int b  = task / TILES;
  const int i0 = (task % TILES) * 16;

  float* wbase = smem + wid * (2 * 3 * 16 * KC);  // 3072 floats per wave

  auto issue = [&](int kc, float* buf) {
#pragma unroll
    for (int j = 0; j < 4; ++j) {
      int t   = lane + 32 * j;      // 128 b128-transfers per stream
      int row = t >> 3;             // 16 rows, 8 x b128 per KC-row
      int c4  = (t & 7) * 4;        // float offset within the row chunk
      int rp  = i0 + row;     if (rp > SS - 1) rp = SS - 1;   // clamp (tail tiles)
      int rb  = i0 + 1 + row; if (rb > SS - 1) rb = SS - 1;
      const float* gp = pred  + ((size_t)(b * SS + rp) * NQ)      + kc + c4;
      const float* g0 = batch + ((size_t)(b * SS + rb) * (2*NQ))  + kc + c4;
      const float* g1 = g0 + NQ;
      async_copy_b128(gp, buf + 0 * (16 * KC) + t * 4);
      async_copy_b128(g0, buf + 1 * (16 * KC) + t * 4);
      async_copy_b128(g1, buf + 2 * (16 * KC) + t * 4);
    }
  };

  const int rowA = lane & 15;        // A: M row / B: N column owned by this lane
  const int ksel = (lane >> 4) * 2;  // K split across half-waves per ISA layout

  v8f  acc   = {};
  float dpart = 0.0f;

  issue(0, wbase);
  for (int kc = 0; kc < NQ; kc += KC) {
    float* cur = wbase + (((kc / KC) & 1) ? (3 * 16 * KC) : 0);
    float* nxt = wbase + (((kc / KC) & 1) ? 0 : (3 * 16 * KC));
    if (kc + KC < NQ) { issue(kc + KC, nxt); wait_asynccnt<12>(); }  // prev 12 done
    else              { wait_asynccnt<0>(); }

    const float* pb = cur;                // pred rows
    const float* h0 = cur + 16 * KC;      // batch[:, :Q] rows
    const float* h1 = cur + 32 * KC;      // batch[:, Q:] rows
#pragma unroll
    for (int k4 = 0; k4 < KC; k4 += 4) {
      int off = rowA * KC + k4 + ksel;    // 8-byte aligned
      v2f av = *(const v2f*)(pb + off);
      v2f x0 = *(const v2f*)(h0 + off);
      v2f x1 = *(const v2f*)(h1 + off);
      v2f bv; bv.x = x0.x + x1.x; bv.y = x0.y + x1.y;        // delta
      dpart += (x0.x - x1.x) + (x0.y - x1.y);                // diff partial
      acc = __builtin_amdgcn_wmma_f32_16x16x4_f32(
          false, av, false, bv, (short)0, acc, false, false);
    }
  }

  // diff row L lives half in lane L, half in lane L+16
  float dfull = dpart + __shfl_xor(dpart, 16, 32);
  float aval  = floorf((dfull + 1.0f) * 0.5f);

  // diag(C): C[i][i] -> lane i (vgpr i) for i<8; lane 16+i (vgpr i-8) for i>=8
  int iloc = -1, j = 0;
  if (lane < 8)        { iloc = lane;      j = lane;      }
  else if (lane >= 24) { iloc = lane - 16; j = lane - 24; }
  if (iloc >= 0) {
    int i = i0 + iloc;
    if (i < SM1) {
      p_ws[b * SM1 + i] = acc[j];
      a_ws[b * SM1 + i] = aval;
    }
  }
}

// ---------------- Phase 2: per-student scan + BCE (deterministic) ----------------
__global__ __launch_bounds__(256) void dkt_phase2(
    const float* __restrict__ p_ws, const float* __restrict__ a_ws,
    float* __restrict__ out, float* __restrict__ loss_ws) {
  __shared__ int   si[256];
  __shared__ float sf[256];
  __shared__ int   sc[256];
  const int b = blockIdx.x;
  const int t = threadIdx.x;
  const bool act = (t < SM1);
  float p = act ? p_ws[b * SM1 + t] : 0.0f;
  float a = act ? a_ws[b * SM1 + t] : 0.0f;

  // last index with p > 0 (fixed-order tree max)
  si[t] = (act && p > 0.0f) ? t : -1;
  __syncthreads();
  for (int s = 128; s > 0; s >>= 1) {
    if (t < s) si[t] = max(si[t], si[t + s]);
    __syncthreads();
  }
  const int last = si[0];
  __syncthreads();

  const bool valid = act && ((last < 0) || (t <= last));
  const float vf = valid ? 1.0f : 0.0f;
  if (act) {
    out[1 + b * SM1 + t]             = p * vf;   // prediction slab
    out[1 + BB * SM1 + b * SM1 + t]  = a * vf;   // ground_truth slab
  }

  float lp  = fmaxf(logf(p), -100.0f);          // torch-style log clamp
  float l1p = fmaxf(logf(1.0f - p), -100.0f);
  float bce = valid ? -(a * lp + (1.0f - a) * l1p) : 0.0f;

  sf[t] = bce; sc[t] = valid ? 1 : 0;
  __syncthreads();
  for (int s = 128; s > 0; s >>= 1) {
    if (t < s) { sf[t] += sf[t + s]; sc[t] += sc[t + s]; }
    __syncthreads();
  }
  if (t == 0) loss_ws[b] = sf[0] / (float)sc[0];
}

// ---------------- Phase 3: fixed-order sum of per-student losses ----------------
__global__ __launch_bounds__(256) void dkt_phase3(
    const float* __restrict__ loss_ws, float* __restrict__ out) {
  __shared__ float sf[256];
  const int t = threadIdx.x;
  sf[t] = loss_ws[t];                  // B == 256 exactly
  __syncthreads();
  for (int s = 128; s > 0; s >>= 1) {
    if (t < s) sf[t] += sf[t + s];
    __syncthreads();
  }
  if (t == 0) out[0] = sf[0];
}

extern "C" void kernel_launch(void* const* d_in, const int* in_sizes, int n_in,
                              void* d_out, int out_size, void* d_ws, size_t ws_size,
                              hipStream_t stream) {
  (void)in_sizes; (void)n_in; (void)out_size; (void)ws_size;
  const float* pred  = (const float*)d_in[0];
  const float* batch = (const float*)d_in[1];
  float* out  = (float*)d_out;
  float* p_ws = (float*)d_ws;            // [B*199]
  float* a_ws = p_ws + BB * SM1;         // [B*199]
  float* l_ws = a_ws + BB * SM1;         // [B]

  const int tasks  = BB * TILES;
  const int blocks = (tasks + WPB - 1) / WPB;
  dkt_phase1<<<blocks, WPB * 32, 0, stream>>>(pred, batch, p_ws, a_ws);
  dkt_phase2<<<BB, 256, 0, stream>>>(p_ws, a_ws, out, l_ws);
  dkt_phase3<<<1, 256, 0, stream>>>(l_ws, out);
}